// MLP_Attention_1580547968966
// MI455X (gfx1250) — compile-verified
//
#include <hip/hip_runtime.h>
#include <hip/hip_bf16.h>
#include <math.h>

// Problem sizes (fixed by the reference)
#define BQ  8
#define NXQ 512
#define NRQ 512
#define DQ  256
#define HQ  256

typedef float v2f __attribute__((ext_vector_type(2)));
typedef float v8f __attribute__((ext_vector_type(8)));

__device__ __forceinline__ float fast_tanh(float x) {
#if __has_builtin(__builtin_amdgcn_tanhf)
    return __builtin_amdgcn_tanhf(x);       // v_tanh_f32 on gfx1250
#elif __has_builtin(__builtin_amdgcn_tanh_f32)
    return __builtin_amdgcn_tanh_f32(x);
#else
    return tanhf(x);
#endif
}

__device__ __forceinline__ v8f addsplat(v8f v, float s) {
    v8f r;
#pragma unroll
    for (int i = 0; i < 8; ++i) r[i] = v[i] + s;
    return r;
}

// CDNA5 async copy global -> LDS (tracked by ASYNCcnt, CDNA5 ISA).
// %0 = LDS byte address (VGPR), %1 = 64-bit global address (VGPR pair).
#define ASYNC_G2L_B128(ldsaddr, gptr)                                          \
    asm volatile("global_load_async_to_lds_b128 %0, %1, off"                   \
                 :: "v"(ldsaddr), "v"(gptr) : "memory")
#define WAIT_ASYNCCNT_0() asm volatile("s_wait_asynccnt 0x0" ::: "memory")

// LDS byte offset of a flat pointer to __shared__ (addr[31:0] per ISA aperture map)
__device__ __forceinline__ unsigned lds_off(const void* p) {
    return (unsigned)(unsigned long long)(uintptr_t)p;
}

// -----------------------------------------------------------------------------
// Generic fp32 WMMA GEMM:  C[M,N] = A[M,K] @ B[K,N] (+ bias[n])
//   BMODE 0: B given as W[N,K] row-major  (B[k][n] = W[n*ldb + k])  -> float2 loads
//   BMODE 1: B given as  [K,N] row-major  (B[k][n] = Bm[k*ldb + n]) -> two b32 loads
// One wave computes a 16(M) x 64(N) tile with 4 accumulators, K-step 4
// (V_WMMA_F32_16X16X4_F32), with 2-deep register double-buffering so the next
// k-step's loads are in flight while the current 4 WMMAs execute.
// Per-lane A/B layout (ISA 7.12.2):
//   element i of float2 = Mat[row/col = lane%16][k = k0 + 2*(lane/16) + i]
// -----------------------------------------------------------------------------
template <int BMODE>
__device__ __forceinline__ void gemm_load(const float* __restrict__ arow,
                                          const float* __restrict__ Bb,
                                          int ldb, int n0, int hi, int k,
                                          v2f& a, v2f& b0, v2f& b1, v2f& b2, v2f& b3)
{
    a = *(const v2f*)(arow + k);
    if (BMODE == 0) {
        const size_t ko = (size_t)(k + 2 * hi);
        b0 = *(const v2f*)(Bb + (size_t)(n0 +  0) * ldb + ko);
        b1 = *(const v2f*)(Bb + (size_t)(n0 + 16) * ldb + ko);
        b2 = *(const v2f*)(Bb + (size_t)(n0 + 32) * ldb + ko);
        b3 = *(const v2f*)(Bb + (size_t)(n0 + 48) * ldb + ko);
    } else {
        const float* br0 = Bb + (size_t)(k + 2 * hi) * ldb;
        const float* br1 = br0 + ldb;
        b0.x = br0[n0 +  0]; b0.y = br1[n0 +  0];
        b1.x = br0[n0 + 16]; b1.y = br1[n0 + 16];
        b2.x = br0[n0 + 32]; b2.y = br1[n0 + 32];
        b3.x = br0[n0 + 48]; b3.y = br1[n0 + 48];
    }
}

#define WMMA_QUAD(aa, bb0, bb1, bb2, bb3)                                                           \
    do {                                                                                            \
        acc0 = __builtin_amdgcn_wmma_f32_16x16x4_f32(false, aa, false, bb0, (short)0, acc0, false, false); \
        acc1 = __builtin_amdgcn_wmma_f32_16x16x4_f32(false, aa, false, bb1, (short)0, acc1, false, false); \
        acc2 = __builtin_amdgcn_wmma_f32_16x16x4_f32(false, aa, false, bb2, (short)0, acc2, false, false); \
        acc3 = __builtin_amdgcn_wmma_f32_16x16x4_f32(false, aa, false, bb3, (short)0, acc3, false, false); \
    } while (0)

template <int BMODE>
__global__ void wmma_gemm_f32(const float* __restrict__ A,
                              const float* __restrict__ Bm,
                              const float* __restrict__ bias,
                              float* __restrict__ C,
                              int M, int N, int K,
                              int lda, int ldb, int ldc,
                              long long sA, long long sB, long long sC)
{
    const int wave = threadIdx.x >> 5;
    const int lane = threadIdx.x & 31;
    const int hi   = lane >> 4;    // 0: K pair {0,1}, 1: K pair {2,3}
    const int ln   = lane & 15;    // row (A) / col (B,C)

    const int tilesN = N >> 6;
    const int tile   = blockIdx.x * (blockDim.x >> 5) + wave;
    const int mT     = tile / tilesN;
    const int nT     = tile % tilesN;
    if (mT * 16 >= M) return;

    const float* Ab = A  + (size_t)blockIdx.z * sA;
    const float* Bb = Bm + (size_t)blockIdx.z * sB;
    float*       Cb = C  + (size_t)blockIdx.z * sC;

    v8f acc0 = {}, acc1 = {}, acc2 = {}, acc3 = {};

    const int m  = mT * 16 + ln;
    const int n0 = nT * 64 + ln;
    const float* arow = Ab + (size_t)m * lda + 2 * hi;

    // Software-pipelined K loop: K/4 steps (K is a multiple of 8 here).
    v2f aA, b0A, b1A, b2A, b3A;
    v2f aB, b0B, b1B, b2B, b3B;
    gemm_load<BMODE>(arow, Bb, ldb, n0, hi, 0, aA, b0A, b1A, b2A, b3A);

    const int S = K >> 2;   // even
    for (int s = 0; s < S - 2; s += 2) {
        gemm_load<BMODE>(arow, Bb, ldb, n0, hi, (s + 1) * 4, aB, b0B, b1B, b2B, b3B);
        WMMA_QUAD(aA, b0A, b1A, b2A, b3A);
        gemm_load<BMODE>(arow, Bb, ldb, n0, hi, (s + 2) * 4, aA, b0A, b1A, b2A, b3A);
        WMMA_QUAD(aB, b0B, b1B, b2B, b3B);
    }
    gemm_load<BMODE>(arow, Bb, ldb, n0, hi, (S - 1) * 4, aB, b0B, b1B, b2B, b3B);
    WMMA_QUAD(aA, b0A, b1A, b2A, b3A);
    WMMA_QUAD(aB, b0B, b1B, b2B, b3B);

    if (bias) {
        acc0 = addsplat(acc0, bias[n0 +  0]);
        acc1 = addsplat(acc1, bias[n0 + 16]);
        acc2 = addsplat(acc2, bias[n0 + 32]);
        acc3 = addsplat(acc3, bias[n0 + 48]);
    }

    // C/D layout: VGPR v, lane -> M = mT*16 + v + 8*(lane/16), N = n0 + 16*j
#pragma unroll
    for (int v = 0; v < 8; ++v) {
        const int mm = mT * 16 + v + 8 * hi;
        float* crow = Cb + (size_t)mm * ldc;
        crow[n0 +  0] = acc0[v];
        crow[n0 + 16] = acc1[v];
        crow[n0 + 32] = acc2[v];
        crow[n0 + 48] = acc3[v];
    }
}

// -----------------------------------------------------------------------------
// Score kernel: S[b][r][x] = v_b + sum_h v[h] * tanh(hX[b,x,h] + hR[b,r,h])
// Block = 256 threads handles a 32(x) x 32(r) tile for one batch; h is
// processed in two 128-chunks. Tiles are staged into LDS with the CDNA5
// async-to-LDS engine (global_load_async_to_lds_b128 + s_wait_asynccnt).
// hX tile padded to stride 132 floats (528 B: keeps b128 16-byte LDS
// alignment; 132 % 64 == 4 -> worst case 2-way bank conflict). hR reads are
// wave-broadcast (all lanes of a wave share rb), so no padding needed.
// -----------------------------------------------------------------------------
#define SXS 132

__global__ void score_kernel(const float* __restrict__ hX,
                             const float* __restrict__ hR,
                             const float* __restrict__ vw,
                             const float* __restrict__ vb,
                             float* __restrict__ S)
{
    __shared__ float sx[32 * SXS];
    __shared__ float sr[32 * 128];
    __shared__ float sv[HQ];

    const int b   = blockIdx.z;
    const int x0  = blockIdx.x * 32;
    const int r0  = blockIdx.y * 32;
    const int tid = threadIdx.x;            // 256 threads

    const float* gx = hX + ((size_t)b * NXQ + x0) * HQ;
    const float* gr = hR + ((size_t)b * NRQ + r0) * HQ;

    sv[tid] = vw[tid];

    const int xi = tid & 31;
    const int rb = (tid >> 5) * 4;          // this thread's 4 r-rows

    float acc0 = 0.f, acc1 = 0.f, acc2 = 0.f, acc3 = 0.f;

    for (int hc = 0; hc < HQ; hc += 128) {
        __syncthreads();                    // all readers of the previous chunk done
        // async-stage 32x128 hX and hR sub-tiles: 4 floats (b128) per thread per pass
        for (int i = tid * 4; i < 32 * 128; i += 256 * 4) {
            const int row = i >> 7, col = i & 127;
            const float* gxp = gx + (size_t)row * HQ + hc + col;
            const float* grp = gr + (size_t)row * HQ + hc + col;
            ASYNC_G2L_B128(lds_off(sx + row * SXS + col), gxp);
            ASYNC_G2L_B128(lds_off(sr + row * 128 + col), grp);
        }
        // prefetch next h-chunk into L2 while we compute on this one
        if (hc + 128 < HQ) {
            __builtin_prefetch(gx + (size_t)(tid >> 3) * HQ + hc + 128 + (tid & 7) * 16, 0, 1);
            __builtin_prefetch(gr + (size_t)(tid >> 3) * HQ + hc + 128 + (tid & 7) * 16, 0, 1);
        }
        WAIT_ASYNCCNT_0();
        __syncthreads();

        const float* px  = sx + xi * SXS;
        const float* pr0 = sr + (rb + 0) * 128;
        const float* pr1 = sr + (rb + 1) * 128;
        const float* pr2 = sr + (rb + 2) * 128;
        const float* pr3 = sr + (rb + 3) * 128;
        const float* pv  = sv + hc;
#pragma unroll 4
        for (int h = 0; h < 128; ++h) {
            const float a = px[h];
            const float v = pv[h];
            acc0 += v * fast_tanh(a + pr0[h]);
            acc1 += v * fast_tanh(a + pr1[h]);
            acc2 += v * fast_tanh(a + pr2[h]);
            acc3 += v * fast_tanh(a + pr3[h]);
        }
    }

    const float bb = vb[0];
    float* p = S + ((size_t)b * NRQ + r0 + rb) * NXQ + x0 + xi;
    p[0 * NXQ] = acc0 + bb;
    p[1 * NXQ] = acc1 + bb;
    p[2 * NXQ] = acc2 + bb;
    p[3 * NXQ] = acc3 + bb;
}

// -----------------------------------------------------------------------------
// Softmax over a contiguous row of 512 floats (softmax over x for each (b,r)).
// One block of 256 threads per row.
// -----------------------------------------------------------------------------
__global__ void softmax512(float* __restrict__ S)
{
    __shared__ float red[256];
    const int tid = threadIdx.x;
    float* p = S + (size_t)blockIdx.x * 512;

    const float a = p[tid];
    const float c = p[tid + 256];

    red[tid] = fmaxf(a, c);
    __syncthreads();
#pragma unroll
    for (int s = 128; s > 0; s >>= 1) {
        if (tid < s) red[tid] = fmaxf(red[tid], red[tid + s]);
        __syncthreads();
    }
    const float m = red[0];
    __syncthreads();

    const float e0 = __expf(a - m);
    const float e1 = __expf(c - m);
    red[tid] = e0 + e1;
    __syncthreads();
#pragma unroll
    for (int s = 128; s > 0; s >>= 1) {
        if (tid < s) red[tid] += red[tid + s];
        __syncthreads();
    }
    const float inv = 1.0f / red[0];

    p[tid]       = e0 * inv;
    p[tid + 256] = e1 * inv;
}

// -----------------------------------------------------------------------------
// Launcher.
// Inputs: X, ref, W_X, b_X, W_ref, b_ref, v_w, v_b (all fp32)
// Workspace layout: hX (4 MB) | hR (4 MB) | S (8 MB)  = 16 MB
// -----------------------------------------------------------------------------
extern "C" void kernel_launch(void* const* d_in, const int* in_sizes, int n_in,
                              void* d_out, int out_size, void* d_ws, size_t ws_size,
                              hipStream_t stream)
{
    (void)in_sizes; (void)n_in; (void)out_size; (void)ws_size;

    const float* X     = (const float*)d_in[0];
    const float* ref   = (const float*)d_in[1];
    const float* W_X   = (const float*)d_in[2];
    const float* b_X   = (const float*)d_in[3];
    const float* W_ref = (const float*)d_in[4];
    const float* b_ref = (const float*)d_in[5];
    const float* v_w   = (const float*)d_in[6];
    const float* v_b   = (const float*)d_in[7];
    float*       out   = (float*)d_out;

    float* hX = (float*)d_ws;
    float* hR = hX + (size_t)BQ * NXQ * HQ;
    float* S  = hR + (size_t)BQ * NRQ * HQ;

    // hX = X @ W_X^T + b_X   (M = B*NX = 4096, N = H = 256, K = D = 256)
    // wave tiles = (4096/16)*(256/64) = 1024 -> 128 blocks of 8 waves
    wmma_gemm_f32<0><<<dim3(128, 1, 1), 256, 0, stream>>>(
        X, W_X, b_X, hX, BQ * NXQ, HQ, DQ, DQ, DQ, HQ, 0, 0, 0);

    // hR = ref @ W_ref^T + b_ref
    wmma_gemm_f32<0><<<dim3(128, 1, 1), 256, 0, stream>>>(
        ref, W_ref, b_ref, hR, BQ * NRQ, HQ, DQ, DQ, DQ, HQ, 0, 0, 0);

    // S[b][r][x] = v_b + sum_h v_w[h] * tanh(hX + hR)
    score_kernel<<<dim3(NXQ / 32, NRQ / 32, BQ), 256, 0, stream>>>(
        hX, hR, v_w, v_b, S);

    // softmax over x (contiguous rows of 512)
    softmax512<<<dim3(BQ * NRQ, 1, 1), 256, 0, stream>>>(S);

    // out[b] = S[b] @ X[b]   (M = NR = 512, N = D = 256, K = NX = 512)
    // wave tiles per batch = (512/16)*(256/64) = 128 -> 16 blocks, z = B
    wmma_gemm_f32<1><<<dim3(16, 1, BQ), 256, 0, stream>>>(
        S, X, nullptr, out, NRQ, DQ, NXQ, NXQ, DQ, DQ,
        (long long)NRQ * NXQ, (long long)NXQ * DQ, (long long)NRQ * DQ);
}